// ST_HGAT_24790551232750
// MI455X (gfx1250) — compile-verified
//
#include <hip/hip_runtime.h>
#include <math.h>
#include <limits.h>

#define HID 64
#define APITCH 68
#define GEMM_ROWS 32

typedef __attribute__((ext_vector_type(2))) float v2f;
typedef __attribute__((ext_vector_type(8))) float v8f;

// ---------------- helpers ----------------
__device__ __forceinline__ int f32_ord(float f) {
    int i = __float_as_int(f);
    return (i >= 0) ? i : (i ^ 0x7FFFFFFF);
}
__device__ __forceinline__ float ord_f32(int i) {
    return __int_as_float((i >= 0) ? i : (i ^ 0x7FFFFFFF));
}

// ---------------- fill kernels ----------------
__global__ void fill_f32_kernel(float* p, float v, int n) {
    int i = blockIdx.x * blockDim.x + threadIdx.x;
    if (i < n) p[i] = v;
}
__global__ void fill_i32_kernel(int* p, int v, int n) {
    int i = blockIdx.x * blockDim.x + threadIdx.x;
    if (i < n) p[i] = v;
}

// ---------------- input projection: h = elu(x @ W + b) ----------------
__global__ void proj_kernel(const float* __restrict__ x, const float* __restrict__ W,
                            const float* __restrict__ b, float* __restrict__ h,
                            int n, int din) {
    int i = blockIdx.x * blockDim.x + threadIdx.x;
    int node = i >> 6, c = i & 63;
    if (node >= n) return;
    float acc = b[c];
    for (int k = 0; k < din; ++k) acc += x[node * din + k] * W[k * HID + c];
    h[i] = acc > 0.f ? acc : expm1f(acc);
}

// ---------------- Y[N,64] = X[N,64] @ W[64,64] + bias, via V_WMMA_F32_16X16X4_F32 ----------------
// Block: 256 threads = 8 waves. Block covers 32 rows; wave w -> (mtile = w/4, ntile = w%4).
__global__ __launch_bounds__(256) void gemm64_wmma_kernel(
    const float* __restrict__ X, const float* __restrict__ W,
    const float* __restrict__ bias, float* __restrict__ Y, int N)
{
    __shared__ float As[GEMM_ROWS * APITCH];
    __shared__ float Ws[64 * APITCH];
    int tid = threadIdx.x;
    int r0 = blockIdx.x * GEMM_ROWS;

    for (int idx = tid; idx < 64 * 64; idx += 256) {
        int r = idx >> 6, c = idx & 63;
        Ws[r * APITCH + c] = W[idx];
    }
    for (int idx = tid; idx < GEMM_ROWS * 64; idx += 256) {
        int r = idx >> 6, c = idx & 63;
        As[r * APITCH + c] = (r0 + r < N) ? X[(size_t)(r0 + r) * 64 + c] : 0.f;
    }
    __syncthreads();

    int wave = tid >> 5;
    int lane = tid & 31;
    int mt = wave >> 2;          // 0..1
    int nt = wave & 3;           // 0..3
    int lmod = lane & 15;
    int hi = lane >> 4;          // lane half selects K pair {0,1} vs {2,3}

    const float* arow = &As[(mt * 16 + lmod) * APITCH];
    const float* bcol = &Ws[nt * 16 + lmod];

    v8f acc = {};
    for (int s = 0; s < 16; ++s) {
        int k0 = 4 * s + 2 * hi;
        v2f a, b;
        a.x = arow[k0];
        a.y = arow[k0 + 1];
        b.x = bcol[(size_t)k0 * APITCH];
        b.y = bcol[(size_t)(k0 + 1) * APITCH];
        acc = __builtin_amdgcn_wmma_f32_16x16x4_f32(false, a, false, b,
                                                    (short)0, acc, false, false);
    }

    int colg = nt * 16 + lmod;
    float bv = bias[colg];
    for (int i = 0; i < 8; ++i) {
        int m = hi ? (i + 8) : i;          // C/D layout: VGPR i -> M=i (lanes 0-15) / M=i+8
        int rowg = r0 + mt * 16 + m;
        if (rowg < N) Y[(size_t)rowg * 64 + colg] = acc[i] + bv;
    }
}

// ---------------- edge pass A: logits + atomic max per (dst, head) ----------------
// one wave per edge; lane handles channels c=lane and c=lane+32
__global__ __launch_bounds__(256) void edge_logits_kernel(
    const float* __restrict__ XL, const float* __restrict__ XR,
    const float* __restrict__ ea, const float* __restrict__ We,
    const float* __restrict__ att, const int* __restrict__ src,
    const int* __restrict__ dst, float* __restrict__ logits,
    int* __restrict__ mx, int E)
{
    __shared__ float WeS[4 * 64];
    __shared__ float attS[64];
    int tid = threadIdx.x;
    WeS[tid] = We[tid];
    if (tid < 64) attS[tid] = att[tid];
    __syncthreads();

    int e = blockIdx.x * 8 + (tid >> 5);
    if (e >= E) return;
    int lane = tid & 31;
    int s = src[e], d = dst[e];
    float e0 = ea[e * 4 + 0], e1 = ea[e * 4 + 1], e2 = ea[e * 4 + 2], e3 = ea[e * 4 + 3];

    float r0, r1;
    {
        int c = lane;
        float ep = e0 * WeS[c] + e1 * WeS[64 + c] + e2 * WeS[128 + c] + e3 * WeS[192 + c];
        float m = XL[(size_t)s * 64 + c] + XR[(size_t)d * 64 + c] + ep;
        m = m > 0.f ? m : 0.2f * m;
        r0 = m * attS[c];
    }
    {
        int c = lane + 32;
        float ep = e0 * WeS[c] + e1 * WeS[64 + c] + e2 * WeS[128 + c] + e3 * WeS[192 + c];
        float m = XL[(size_t)s * 64 + c] + XR[(size_t)d * 64 + c] + ep;
        m = m > 0.f ? m : 0.2f * m;
        r1 = m * attS[c];
    }
    // reduce within 16-lane halves -> head sums
    for (int off = 8; off >= 1; off >>= 1) {
        r0 += __shfl_xor(r0, off, 32);
        r1 += __shfl_xor(r1, off, 32);
    }
    if ((lane & 15) == 0) {
        int h0 = lane >> 4;                 // lane0 -> heads {0,2}; lane16 -> heads {1,3}
        logits[e * 4 + h0]     = r0;
        logits[e * 4 + h0 + 2] = r1;
        atomicMax(&mx[d * 4 + h0],     f32_ord(r0));
        atomicMax(&mx[d * 4 + h0 + 2], f32_ord(r1));
    }
}

// ---------------- edge pass B: a = exp(logit - mx[dst]); den += a ----------------
__global__ void edge_exp_kernel(const int* __restrict__ dst, float* __restrict__ logits,
                                const int* __restrict__ mx, float* __restrict__ den, int E4)
{
    int i = blockIdx.x * blockDim.x + threadIdx.x;
    if (i >= E4) return;
    int e = i >> 2, h = i & 3;
    int d = dst[e];
    float m = ord_f32(mx[d * 4 + h]);
    float a = expf(logits[i] - m);
    logits[i] = a;
    atomicAdd(&den[d * 4 + h], a);
}

// ---------------- edge pass C: acc[dst] += (a/den) * XL[src] ----------------
__global__ __launch_bounds__(256) void edge_aggr_kernel(
    const float* __restrict__ XL, const float* __restrict__ aval,
    const float* __restrict__ den, const int* __restrict__ src,
    const int* __restrict__ dst, float* __restrict__ acc, int E)
{
    int tid = threadIdx.x;
    int e = blockIdx.x * 8 + (tid >> 5);
    if (e >= E) return;
    int lane = tid & 31;
    int s = src[e], d = dst[e];
    int hh = lane >> 4;
    float a0 = aval[e * 4 + hh]     / (den[d * 4 + hh]     + 1e-16f);
    float a1 = aval[e * 4 + hh + 2] / (den[d * 4 + hh + 2] + 1e-16f);
    atomicAdd(&acc[(size_t)d * 64 + lane],      a0 * XL[(size_t)s * 64 + lane]);
    atomicAdd(&acc[(size_t)d * 64 + lane + 32], a1 * XL[(size_t)s * 64 + lane + 32]);
}

// ---------------- h = elu(acc + bias0 (+ bias1)) ----------------
__global__ void elu_bias_kernel(const float* __restrict__ acc, const float* __restrict__ b0,
                                const float* __restrict__ b1, float* __restrict__ h, int n64)
{
    int i = blockIdx.x * blockDim.x + threadIdx.x;
    if (i >= n64) return;
    int c = i & 63;
    float x = acc[i] + b0[c] + (b1 ? b1[c] : 0.f);
    h[i] = x > 0.f ? x : expm1f(x);
}

// ---------------- batchnorm stats: stats[0:64]=sum, stats[64:128]=sumsq ----------------
__global__ __launch_bounds__(256) void bn_stats_kernel(const float* __restrict__ h,
                                                       float* __restrict__ stats, int n)
{
    __shared__ float sS[256], ssS[256];
    int tid = threadIdx.x;
    int c = tid & 63;
    int rgrp = tid >> 6;   // 0..3
    float s = 0.f, ss = 0.f;
    for (int r = blockIdx.x * 4 + rgrp; r < n; r += gridDim.x * 4) {
        float v = h[(size_t)r * 64 + c];
        s += v; ss += v * v;
    }
    sS[tid] = s; ssS[tid] = ss;
    __syncthreads();
    if (tid < 64) {
        float st  = sS[tid]  + sS[tid + 64]  + sS[tid + 128]  + sS[tid + 192];
        float sst = ssS[tid] + ssS[tid + 64] + ssS[tid + 128] + ssS[tid + 192];
        atomicAdd(&stats[tid], st);
        atomicAdd(&stats[64 + tid], sst);
    }
}

__global__ void bn_norm_kernel(float* __restrict__ h, const float* __restrict__ stats,
                               const float* __restrict__ g, const float* __restrict__ b, int n)
{
    int i = blockIdx.x * blockDim.x + threadIdx.x;
    if (i >= n * 64) return;
    int c = i & 63;
    float invn = 1.f / (float)n;
    float mu = stats[c] * invn;
    float var = stats[64 + c] * invn - mu * mu;
    h[i] = (h[i] - mu) * rsqrtf(var + 1e-5f) * g[c] + b[c];
}

// ---------------- head: out[n] = dot(h[n], W) + b ----------------
__global__ __launch_bounds__(256) void head_kernel(const float* __restrict__ h,
                                                   const float* __restrict__ W,
                                                   const float* __restrict__ b,
                                                   float* __restrict__ out, int n)
{
    int tid = threadIdx.x;
    int node = blockIdx.x * 8 + (tid >> 5);
    if (node >= n) return;
    int lane = tid & 31;
    float s = h[(size_t)node * 64 + lane] * W[lane]
            + h[(size_t)node * 64 + lane + 32] * W[lane + 32];
    for (int off = 16; off >= 1; off >>= 1) s += __shfl_xor(s, off, 32);
    if (lane == 0) out[node] = s + b[0];
}

// =====================================================================
extern "C" void kernel_launch(void* const* d_in, const int* in_sizes, int n_in,
                              void* d_out, int out_size, void* d_ws, size_t ws_size,
                              hipStream_t stream)
{
    (void)n_in; (void)out_size; (void)ws_size;
    const float* x_l   = (const float*)d_in[0];
    const float* x_r1  = (const float*)d_in[1];
    const float* x_r2  = (const float*)d_in[2];
    const int*   ll_s  = (const int*)d_in[3];
    const int*   ll_d  = (const int*)d_in[4];
    const float* ea_ll = (const float*)d_in[5];
    const int*   r1_s  = (const int*)d_in[6];
    const int*   r1_d  = (const int*)d_in[7];
    const float* ea_r1 = (const float*)d_in[8];
    const int*   r2_s  = (const int*)d_in[9];
    const int*   r2_d  = (const int*)d_in[10];
    const float* ea_r2 = (const float*)d_in[11];
    const int*   x1_s  = (const int*)d_in[12];
    const int*   x1_d  = (const int*)d_in[13];
    const float* ea_x1 = (const float*)d_in[14];
    const int*   x2_s  = (const int*)d_in[15];
    const int*   x2_d  = (const int*)d_in[16];
    const float* ea_x2 = (const float*)d_in[17];

    const int NL  = in_sizes[0] / 9;
    const int NR  = in_sizes[1] / 7;
    const int Ell = in_sizes[3];
    const int Err = in_sizes[6];
    const int Ex  = in_sizes[12];

    // params: jax pytree order (dict keys sorted): head_lidar, head_radar, layers, proj
    const float* head_l_W = (const float*)d_in[18];
    const float* head_l_b = (const float*)d_in[19];
    const float* head_r_W = (const float*)d_in[20];
    const float* head_r_b = (const float*)d_in[21];

    struct ConvP { const float *We, *Wl, *Wr, *att, *bias, *bl, *br; };
    struct BnP   { const float *b, *g; };
    BnP   bn[2][3];
    ConvP cv[2][5];  // key-sorted: ll, r1, r2, x1, x2
    int idx = 22;
    for (int L = 0; L < 2; ++L) {
        for (int t = 0; t < 3; ++t) {     // bn: {b, g} (sorted)
            bn[L][t].b = (const float*)d_in[idx++];
            bn[L][t].g = (const float*)d_in[idx++];
        }
        for (int k = 0; k < 5; ++k) {     // conv: We, Wl, Wr, att, bias, bl, br (sorted)
            cv[L][k].We   = (const float*)d_in[idx++];
            cv[L][k].Wl   = (const float*)d_in[idx++];
            cv[L][k].Wr   = (const float*)d_in[idx++];
            cv[L][k].att  = (const float*)d_in[idx++];
            cv[L][k].bias = (const float*)d_in[idx++];
            cv[L][k].bl   = (const float*)d_in[idx++];
            cv[L][k].br   = (const float*)d_in[idx++];
        }
    }
    const float* proj_l_W  = (const float*)d_in[idx++];
    const float* proj_l_b  = (const float*)d_in[idx++];
    const float* proj_r1_W = (const float*)d_in[idx++];
    const float* proj_r1_b = (const float*)d_in[idx++];
    const float* proj_r2_W = (const float*)d_in[idx++];
    const float* proj_r2_b = (const float*)d_in[idx++];

    // workspace carve-up
    float* ws = (float*)d_ws;
    size_t off = 0;
    auto carve = [&](size_t nf) { float* p = ws + off; off += nf; return p; };
    float* hl    = carve((size_t)NL * 64);
    float* h1    = carve((size_t)NR * 64);
    float* h2    = carve((size_t)NR * 64);
    float* accl  = carve((size_t)NL * 64);
    float* acc1  = carve((size_t)NR * 64);
    float* acc2  = carve((size_t)NR * 64);
    float* XL    = carve((size_t)NL * 64);
    float* XR    = carve((size_t)NL * 64);
    float* elog  = carve((size_t)Ell * 4);
    float* den   = carve((size_t)NL * 4);
    int*   mx    = (int*)carve((size_t)NL * 4);
    float* stats = carve(128);

    auto fillf = [&](float* p, float v, int n) {
        fill_f32_kernel<<<(n + 255) / 256, 256, 0, stream>>>(p, v, n);
    };
    auto filli = [&](int* p, int v, int n) {
        fill_i32_kernel<<<(n + 255) / 256, 256, 0, stream>>>(p, v, n);
    };

    auto run_conv = [&](const ConvP& p, const float* hsrc, int nsrc,
                        const float* hdst, int ndst,
                        const int* es, const int* ed, const float* ea, int E,
                        float* acc) {
        filli(mx, INT_MIN, ndst * 4);
        fillf(den, 0.f, ndst * 4);
        gemm64_wmma_kernel<<<(nsrc + GEMM_ROWS - 1) / GEMM_ROWS, 256, 0, stream>>>(
            hsrc, p.Wl, p.bl, XL, nsrc);
        gemm64_wmma_kernel<<<(ndst + GEMM_ROWS - 1) / GEMM_ROWS, 256, 0, stream>>>(
            hdst, p.Wr, p.br, XR, ndst);
        edge_logits_kernel<<<(E + 7) / 8, 256, 0, stream>>>(
            XL, XR, ea, p.We, p.att, es, ed, elog, mx, E);
        edge_exp_kernel<<<(E * 4 + 255) / 256, 256, 0, stream>>>(ed, elog, mx, den, E * 4);
        edge_aggr_kernel<<<(E + 7) / 8, 256, 0, stream>>>(XL, elog, den, es, ed, acc, E);
    };

    // input projections
    proj_kernel<<<((size_t)NL * 64 + 255) / 256, 256, 0, stream>>>(x_l,  proj_l_W,  proj_l_b,  hl, NL, 9);
    proj_kernel<<<((size_t)NR * 64 + 255) / 256, 256, 0, stream>>>(x_r1, proj_r1_W, proj_r1_b, h1, NR, 7);
    proj_kernel<<<((size_t)NR * 64 + 255) / 256, 256, 0, stream>>>(x_r2, proj_r2_W, proj_r2_b, h2, NR, 7);

    const float* bng[3]; const float* bnb[3];
    for (int L = 0; L < 2; ++L) {
        fillf(accl, 0.f, NL * 64);
        fillf(acc1, 0.f, NR * 64);
        fillf(acc2, 0.f, NR * 64);

        run_conv(cv[L][0], hl, NL, hl, NL, ll_s, ll_d, ea_ll, Ell, accl);  // ll
        run_conv(cv[L][1], h1, NR, h1, NR, r1_s, r1_d, ea_r1, Err, acc1);  // r1
        run_conv(cv[L][3], hl, NL, h1, NR, x1_s, x1_d, ea_x1, Ex,  acc1);  // x1
        run_conv(cv[L][2], h2, NR, h2, NR, r2_s, r2_d, ea_r2, Err, acc2);  // r2
        run_conv(cv[L][4], hl, NL, h2, NR, x2_s, x2_d, ea_x2, Ex,  acc2);  // x2

        // finalize: bias sum + ELU (overwrites h after ALL convs have consumed it)
        elu_bias_kernel<<<(NL * 64 + 255) / 256, 256, 0, stream>>>(accl, cv[L][0].bias, nullptr,         hl, NL * 64);
        elu_bias_kernel<<<(NR * 64 + 255) / 256, 256, 0, stream>>>(acc1, cv[L][1].bias, cv[L][3].bias,   h1, NR * 64);
        elu_bias_kernel<<<(NR * 64 + 255) / 256, 256, 0, stream>>>(acc2, cv[L][2].bias, cv[L][4].bias,   h2, NR * 64);

        float* hs[3] = { hl, h1, h2 };
        int    ns[3] = { NL, NR, NR };
        for (int t = 0; t < 3; ++t) { bnb[t] = bn[L][t].b; bng[t] = bn[L][t].g; }
        for (int t = 0; t < 3; ++t) {
            fillf(stats, 0.f, 128);
            bn_stats_kernel<<<256, 256, 0, stream>>>(hs[t], stats, ns[t]);
            bn_norm_kernel<<<(ns[t] * 64 + 255) / 256, 256, 0, stream>>>(hs[t], stats, bng[t], bnb[t], ns[t]);
        }
    }

    // heads -> concatenated [NL + 2*NR, 1]
    float* out = (float*)d_out;
    head_kernel<<<(NL + 7) / 8, 256, 0, stream>>>(hl, head_l_W, head_l_b, out, NL);
    head_kernel<<<(NR + 7) / 8, 256, 0, stream>>>(h1, head_r_W, head_r_b, out + NL, NR);
    head_kernel<<<(NR + 7) / 8, 256, 0, stream>>>(h2, head_r_W, head_r_b, out + NL + NR, NR);
}